// csi_embedding_54709293416679
// MI455X (gfx1250) — compile-verified
//
#include <hip/hip_runtime.h>

// y[bt, c*32+s] = x[bt, c] * w[c*32+s]
// B*T = 65536, C = 128, S = 32, OUT = 4096.
// Store-bandwidth-bound (1.07 GB writes, roofline ~47us @ 23.3 TB/s).
// Multiply runs on the matrix unit as a K=1 outer product via
// V_WMMA_F32_16X16X4_F32 (exact f32; FLOPs are free at 0.25 FLOP/byte).
// Each wave owns one channel and loops over 8 bt-tiles so the w-row (B matrix)
// and all address setup are amortized: per tile = 1 load + 2 wmma + 16 stores.

typedef __attribute__((ext_vector_type(2))) float v2f;
typedef __attribute__((ext_vector_type(8))) float v8f;

#define CSI_C       128
#define CSI_S       32
#define CSI_OUT     4096
#define CSI_BT      (512 * 128)
#define WAVES_PER_BLOCK 8
#define TILES_PER_WAVE  8   // 16-row tiles per wave (same channel)

__global__ __launch_bounds__(256) void csi_embedding_wmma_kernel(
    const float* __restrict__ x,   // [BT, C]
    const float* __restrict__ w,   // [C*S] flat
    float* __restrict__ y)         // [BT, OUT]
{
    const int lane = threadIdx.x & 31;
    const int wave = threadIdx.x >> 5;
    const int gw   = blockIdx.x * WAVES_PER_BLOCK + wave;   // global wave id

    const int c      = gw & (CSI_C - 1);                    // channel
    const int btbase = (gw >> 7) * (16 * TILES_PER_WAVE);   // first bt row

    const int  half = lane >> 4;      // 0: lanes 0-15, 1: lanes 16-31
    const int  l16  = lane & 15;
    const bool lo   = (lane < 16);

    // ---- B matrices (4x16): only row K=0 nonzero = w[c*S + s0 + n].
    // Layout: VGPR0 = {lanes0-15: K=0 (N=lane), lanes16-31: K=2}, VGPR1 = {K=1, K=3}.
    // All lanes load (lanes>=16 reuse lane&15's address -> same cacheline), then
    // value-select zero so EXEC stays all-ones (required by WMMA).
    const float* wrow = w + c * CSI_S;
    const float  wv0  = wrow[l16];
    const float  wv1  = wrow[16 + l16];
    v2f B0; B0.x = lo ? wv0 : 0.0f; B0.y = 0.0f;
    v2f B1; B1.x = lo ? wv1 : 0.0f; B1.y = 0.0f;

    // Per-lane pointers; advance 16 rows per tile.
    const float* xp  = x + (size_t)(btbase + l16) * CSI_C + c;
    float*       out = y + (size_t)(btbase + half * 8) * CSI_OUT + c * CSI_S + l16;

    const v8f czero = {};

#pragma unroll
    for (int t = 0; t < TILES_PER_WAVE; ++t) {
        // ---- A matrix (16x4 f32): only column K=0 nonzero = x[bt0+m, c].
        const float xv = *xp;
        v2f A; A.x = lo ? xv : 0.0f; A.y = 0.0f;

        // 8 args: (neg_a, A, neg_b, B, c_mod, C, reuse_a, reuse_b)
        v8f d0 = __builtin_amdgcn_wmma_f32_16x16x4_f32(
            false, A, false, B0, (short)0, czero, false, false);
        v8f d1 = __builtin_amdgcn_wmma_f32_16x16x4_f32(
            false, A, false, B1, (short)0, czero, false, false);

        // ---- Store D tiles. D layout: VGPR v, lane l -> M = v + 8*(l>=16), N = l&15.
        // Rows of y are 16 KB apart; d0+d1 fill the channel's full 128B line per row.
#pragma unroll
        for (int v = 0; v < 8; ++v) {
            out[(size_t)v * CSI_OUT]      = d0[v];
            out[(size_t)v * CSI_OUT + 16] = d1[v];
        }

        xp  += 16 * CSI_C;
        out += (size_t)16 * CSI_OUT;
    }
}

extern "C" void kernel_launch(void* const* d_in, const int* in_sizes, int n_in,
                              void* d_out, int out_size, void* d_ws, size_t ws_size,
                              hipStream_t stream) {
    const float* x = (const float*)d_in[0];   // (512,128,128) f32
    const float* w = (const float*)d_in[1];   // (1,4096) f32
    float* y = (float*)d_out;                 // (512,128,4096) f32

    // Waves = (BT / (16*TILES_PER_WAVE)) * C = 512 * 128 = 65536 -> 8192 blocks.
    const int num_waves  = (CSI_BT / (16 * TILES_PER_WAVE)) * CSI_C;
    const int num_blocks = num_waves / WAVES_PER_BLOCK;

    csi_embedding_wmma_kernel<<<num_blocks, 256, 0, stream>>>(x, w, y);
}